// HessianReg_33595234189776
// MI455X (gfx1250) — compile-verified
//
#include <hip/hip_runtime.h>
#include <hip/hip_bf16.h>
#include <math.h>

// Problem geometry (fixed by the reference): img [N=2, C=1, D=64, H=512, W=512] fp32.
#define NB    2
#define DP    64
#define HH    512
#define WW    512
#define TH    4        // rows of H owned per block per plane
#define DC    16       // d-planes per block chunk
#define WSEG  8        // consecutive w elements per thread
#define PITCH 514      // LDS row pitch in floats: 514 % 64 == 2 -> conflict-free lane map
#define ROWS  (TH + 2) // tile rows incl. +2 halo
#define SLOTF (ROWS * PITCH)
#define NSLOT 4        // 4 slots: prefetch depth 2 (plane d+3 in flight during compute of d)
#define NBLOCKS (NB * (HH / TH) * (DP / DC))   // 2 * 128 * 4 = 1024

typedef float v8f __attribute__((ext_vector_type(8)));
typedef float v2f __attribute__((ext_vector_type(2)));
typedef int   v2i __attribute__((ext_vector_type(2)));

typedef __attribute__((address_space(1))) v2i* glob_v2i_p;
typedef __attribute__((address_space(3))) v2i* lds_v2i_p;

#if defined(__has_builtin)
#if __has_builtin(__builtin_amdgcn_global_load_async_to_lds_b64)
#define HAVE_ASYNC_LDS 1
#endif
#if __has_builtin(__builtin_amdgcn_s_wait_asynccnt)
#define HAVE_WAIT_ASYNC 1
#endif
#endif

__device__ __forceinline__ void async_copy_b64(const float* gsrc, float* ldst) {
#ifdef HAVE_ASYNC_LDS
  __builtin_amdgcn_global_load_async_to_lds_b64(
      (glob_v2i_p)gsrc,   // C-style cast: drops const, retypes to int2, AS1
      (lds_v2i_p)ldst,    // generic -> AS3 int2*
      0, 0);
#else
  float2 v = *(const float2*)gsrc;
  *(float2*)ldst = v;
#endif
}

// Wait until at most `n` of this wave's async-to-LDS loads remain outstanding.
// Async loads complete in order, so n==6 means "everything except the plane
// issued this iteration (6 b64 per thread) has landed".
template <int N>
__device__ __forceinline__ void wait_async_le() {
#ifdef HAVE_ASYNC_LDS
#ifdef HAVE_WAIT_ASYNC
  __builtin_amdgcn_s_wait_asynccnt(N);
#else
  asm volatile("s_wait_asynccnt %0" :: "i"(N) : "memory");
#endif
#endif
}

__global__ __launch_bounds__(256) void hess_main(const float* __restrict__ img,
                                                 float* __restrict__ ws) {
  __shared__ float smem[NSLOT * SLOTF];   // 4 * 6 * 514 * 4B = 49,344 B
  __shared__ float redbuf[8];

  const int tid = threadIdx.x;
  const int bid = blockIdx.x;
  const int n      = bid >> 9;          // / 512
  const int rem    = bid & 511;
  const int htile  = rem >> 2;          // / 4
  const int dchunk = rem & 3;
  const int h0 = htile * TH;
  const int d0 = dchunk * DC;           // multiple of 16 -> d0 % 4 == 0

  // Async-fill one plane tile (6 rows x 512 floats) into an LDS slot.
  auto loadPlane = [&](int plane, int slot) {
    const size_t pbase = (size_t)(n * DP + plane) * (size_t)(HH * WW);
#pragma unroll
    for (int k = 0; k < ROWS; ++k) {
      int grow = h0 + k;
      if (grow > HH - 1) grow = HH - 1;               // clamp halo; guarded at compute
      const float* src = img + pbase + (size_t)grow * WW + tid * 2;
      float* dst = &smem[slot * SLOTF + k * PITCH + tid * 2];
      async_copy_b64(src, dst);
    }
  };

  // Prologue: planes d0, d0+1, d0+2 (d0+2 <= 50 always exists).
  loadPlane(d0,     0);
  loadPlane(d0 + 1, 1);
  loadPlane(d0 + 2, 2);

  const int r  = tid & 3;          // row within tile
  const int g  = tid >> 2;         // column group (0..63)
  const int w0 = g * WSEG;
  const int h  = h0 + r;
  const bool hok1 = (h + 1) < HH;
  const bool hok2 = (h + 2) < HH;
  const bool fullw = (g < 63);     // all of w0..w0+9 in range

  const float W_Z  = 0.5f;                     // CONTIZ^2
  const float W_XY = 2.0f;
  const float W_SZ = 1.41421356237309515f;     // 2*CONTIZ = sqrt(2)

  float acc0 = 0.0f, acc1 = 0.0f;              // two chains to cut add latency

  for (int dd = 0; dd < DC; ++dd) {
    __syncthreads();              // everyone done reading the slot we are about to overwrite
    const int d = d0 + dd;
    const bool has1 = (d + 1) < DP;
    const bool has2 = (d + 2) < DP;
    const bool prefetch = (dd < DC - 1) && (d + 3 < DP);
    if (prefetch) loadPlane(d + 3, (dd + 3) & 3);
    // If we prefetched, only the just-issued 6 loads may remain outstanding;
    // otherwise drain fully (plane d+2 from last iteration must have landed).
    if (prefetch) wait_async_le<6>(); else wait_async_le<0>();
    __syncthreads();              // everyone's async fills visible

    const int s0 = dd & 3, s1 = (dd + 1) & 3, s2 = (dd + 2) & 3;
    const float* A0 = &smem[s0 * SLOTF + r * PITCH + w0];  // plane d,   row h
    const float* A1 = A0 + PITCH;                          // plane d,   row h+1
    const float* A2 = A1 + PITCH;                          // plane d,   row h+2
    const float* B0 = &smem[s1 * SLOTF + r * PITCH + w0];  // plane d+1, row h
    const float* B1 = B0 + PITCH;                          // plane d+1, row h+1
    const float* C0 = &smem[s2 * SLOTF + r * PITCH + w0];  // plane d+2, row h

    float a0[WSEG + 2], a1[WSEG + 1], a2[WSEG], b0[WSEG + 1], b1[WSEG], c0[WSEG];
#pragma unroll
    for (int i = 0; i < WSEG + 2; ++i) a0[i] = A0[i];
#pragma unroll
    for (int i = 0; i < WSEG + 1; ++i) a1[i] = A1[i];
#pragma unroll
    for (int i = 0; i < WSEG; ++i) a2[i] = A2[i];
#pragma unroll
    for (int i = 0; i < WSEG + 1; ++i) b0[i] = B0[i];
#pragma unroll
    for (int i = 0; i < WSEG; ++i) b1[i] = B1[i];
#pragma unroll
    for (int i = 0; i < WSEG; ++i) c0[i] = C0[i];

    if (fullw & hok2 & has2) {
      // Fast interior path: no guards, |x| folds into src modifiers.
#pragma unroll
      for (int i = 0; i < WSEG; ++i) {
        const float dx0 = a0[i] - a0[i + 1];
        const float dy0 = a0[i] - a1[i];
        const float dz0 = a0[i] - b0[i];
        const float xx = dx0 - (a0[i + 1] - a0[i + 2]);
        const float yy = dy0 - (a1[i] - a2[i]);
        const float zz = dz0 - (b0[i] - c0[i]);
        const float xy = dx0 - (a1[i] - a1[i + 1]);
        const float xz = dx0 - (b0[i] - b0[i + 1]);
        const float yz = dy0 - (b0[i] - b1[i]);
        acc0 += fabsf(xx);
        acc1 += fabsf(yy);
        acc0 += W_Z * fabsf(zz);
        acc1 += W_XY * fabsf(xy);
        acc0 += W_SZ * fabsf(xz);
        acc1 += W_SZ * fabsf(yz);
      }
    } else {
      // Boundary path: per-term validity selects (safe vs. uninit LDS pad).
#pragma unroll
      for (int i = 0; i < WSEG; ++i) {
        const int w = w0 + i;
        const bool wok1 = (w + 1) < WW;
        const bool wok2 = (w + 2) < WW;
        const float xx = a0[i] - 2.0f * a0[i + 1] + a0[i + 2];
        const float yy = a0[i] - 2.0f * a1[i] + a2[i];
        const float zz = a0[i] - 2.0f * b0[i] + c0[i];
        const float xy = a0[i] - a0[i + 1] - a1[i] + a1[i + 1];
        const float xz = a0[i] - a0[i + 1] - b0[i] + b0[i + 1];
        const float yz = a0[i] - a1[i] - b0[i] + b1[i];
        acc0 += wok2 ? fabsf(xx) : 0.0f;
        acc1 += hok2 ? fabsf(yy) : 0.0f;
        acc0 += has2 ? W_Z * fabsf(zz) : 0.0f;
        acc1 += (wok1 && hok1) ? W_XY * fabsf(xy) : 0.0f;
        acc0 += (wok1 && has1) ? W_SZ * fabsf(xz) : 0.0f;
        acc1 += (hok1 && has1) ? W_SZ * fabsf(yz) : 0.0f;
      }
    }
  }

  const float acc = acc0 + acc1;

  // --- Wave reduction via WMMA (f32 16x16x4): D = A * ones + 0.
  // Each lane's acc appears exactly once in A, so row-sums of D collapse the wave.
  v2f av = {acc, 0.0f};
  v2f bv = {1.0f, 1.0f};
  v8f cv = {0.0f, 0.0f, 0.0f, 0.0f, 0.0f, 0.0f, 0.0f, 0.0f};
  cv = __builtin_amdgcn_wmma_f32_16x16x4_f32(false, av, false, bv, (short)0, cv,
                                             false, false);
  float t = cv[0] + cv[1] + cv[2] + cv[3] + cv[4] + cv[5] + cv[6] + cv[7];
  // lanes 0-15 hold sum of D rows 0..7, lanes 16-31 hold rows 8..15.
  float wsum = __shfl(t, 0, 32) + __shfl(t, 16, 32);

  const int lane = tid & 31, wid = tid >> 5;
  if (lane == 0) redbuf[wid] = wsum;
  __syncthreads();
  if (tid == 0) {
    float b = 0.0f;
#pragma unroll
    for (int i = 0; i < 8; ++i) b += redbuf[i];
    ws[bid] = b;
  }
}

// Deterministic fixed-order final reduction of the 1024 block partials.
__global__ __launch_bounds__(256) void hess_final(const float* __restrict__ ws,
                                                  float* __restrict__ out) {
  __shared__ float s[256];
  const int t = threadIdx.x;
  float v = ws[t] + ws[t + 256] + ws[t + 512] + ws[t + 768];
  s[t] = v;
  __syncthreads();
  for (int off = 128; off > 0; off >>= 1) {
    if (t < off) s[t] += s[t + off];
    __syncthreads();
  }
  if (t == 0) out[0] = s[0] * (1.0f / (float)(HH * WW));
}

extern "C" void kernel_launch(void* const* d_in, const int* in_sizes, int n_in,
                              void* d_out, int out_size, void* d_ws, size_t ws_size,
                              hipStream_t stream) {
  (void)in_sizes; (void)n_in; (void)out_size; (void)ws_size;
  const float* img = (const float*)d_in[0];
  float* out = (float*)d_out;
  float* ws = (float*)d_ws;                 // needs 1024 floats (4 KB)
  hess_main<<<NBLOCKS, 256, 0, stream>>>(img, ws);
  hess_final<<<1, 256, 0, stream>>>(ws, out);
}